// GraphLaplacianLoss_5634997093002
// MI455X (gfx1250) — compile-verified
//
#include <hip/hip_runtime.h>
#include <math.h>

typedef __attribute__((ext_vector_type(2))) float v2f;
typedef __attribute__((ext_vector_type(8))) float v8f;

#define NPTS 8192
#define DX   512
#define DZ   64
#define KNN  8
#define ROWS_PER_BLK 32
#define COLS_PER_TILE 64
#define AS_STRIDE 516   // 516 % 64 == 4 -> conflict-free strided LDS reads, float4-aligned

__global__ void gl_zero_out(float* o) { o[0] = 0.0f; }

// One wave (32 lanes) per row: n2[i] = sum_k X[i,k]^2
__global__ __launch_bounds__(256) void gl_row_norms(const float* __restrict__ X,
                                                    float* __restrict__ n2) {
    int wave = threadIdx.x >> 5;
    int lane = threadIdx.x & 31;
    int row  = blockIdx.x * 8 + wave;
    const float* xr = X + (size_t)row * DX;
    float s = 0.0f;
    for (int c = lane; c < DX; c += 32) { float v = xr[c]; s += v * v; }
#pragma unroll
    for (int m = 16; m >= 1; m >>= 1) s += __shfl_xor(s, m, 32);
    if (lane == 0) n2[row] = s;
}

// Fused fp32-WMMA distance GEMM + per-row top-8 (self excluded).
// Block: 128 threads (4 waves), owns 32 rows. Per tile: 32 rows x 64 cols.
// Wave w computes cols [cT + 16w, cT + 16w + 16), two 16x16 WMMA tiles (row groups).
__global__ __launch_bounds__(128) void gl_knn_kernel(const float* __restrict__ X,
                                                     const float* __restrict__ n2,
                                                     int* __restrict__ nbr,
                                                     float* __restrict__ dval) {
    __shared__ float As[ROWS_PER_BLK][AS_STRIDE];
    __shared__ float Dt[ROWS_PER_BLK][COLS_PER_TILE + 1];
    __shared__ float sn2[ROWS_PER_BLK];

    const int tid  = threadIdx.x;
    const int wave = tid >> 5;
    const int lane = tid & 31;
    const int half = lane >> 4;   // 0: K pair {0,1}; 1: K pair {2,3}
    const int l    = lane & 15;
    const int r0   = blockIdx.x * ROWS_PER_BLK;

    // Stage A rows (32 x 512 fp32) into LDS, coalesced float4.
    const float4* X4 = reinterpret_cast<const float4*>(X);
    for (int e4 = tid; e4 < ROWS_PER_BLK * (DX / 4); e4 += 128) {
        int row = e4 >> 7;        // / 128
        int c4  = e4 & 127;
        float4 v = X4[(size_t)(r0 + row) * (DX / 4) + c4];
        *reinterpret_cast<float4*>(&As[row][c4 * 4]) = v;
    }
    if (tid < ROWS_PER_BLK) sn2[tid] = n2[r0 + tid];
    __syncthreads();

    float bestV[KNN];
    int   bestI[KNN];
#pragma unroll
    for (int k = 0; k < KNN; ++k) { bestV[k] = 3.0e38f; bestI[k] = 0; }

    for (int cT = 0; cT < NPTS; cT += COLS_PER_TILE) {
        const int gj = cT + wave * 16 + l;                 // this lane's output column
        const float2* Bp = reinterpret_cast<const float2*>(X) + (size_t)gj * (DX / 2);
        const float n2j = n2[gj];

        v8f acc0 = {0.f,0.f,0.f,0.f,0.f,0.f,0.f,0.f};
        v8f acc1 = {0.f,0.f,0.f,0.f,0.f,0.f,0.f,0.f};
#pragma unroll 4
        for (int kb = 0; kb < DX; kb += 4) {
            // B (4x16): lane holds B[K = kb+2*half + {0,1}][N = l] = X[col][k]
            float2 bl = Bp[(kb >> 1) + half];
            v2f b; b.x = bl.x; b.y = bl.y;
            // A (16x4): lane holds A[M = l][K = kb+2*half + {0,1}] for each row group
            float2 a0l = *reinterpret_cast<const float2*>(&As[l][kb + 2 * half]);
            float2 a1l = *reinterpret_cast<const float2*>(&As[16 + l][kb + 2 * half]);
            v2f a0; a0.x = a0l.x; a0.y = a0l.y;
            v2f a1; a1.x = a1l.x; a1.y = a1l.y;
            acc0 = __builtin_amdgcn_wmma_f32_16x16x4_f32(false, a0, false, b,
                                                         (short)0, acc0, false, false);
            acc1 = __builtin_amdgcn_wmma_f32_16x16x4_f32(false, a1, false, b,
                                                         (short)0, acc1, false, false);
        }
        // D layout: VGPR r -> M = r + 8*half, N = l.  Dsq = n2_i + n2_j - 2*dot, clamped.
#pragma unroll
        for (int r = 0; r < 8; ++r) {
            int M = r + 8 * half;
            float d0 = fmaxf(sn2[M]      + n2j - 2.0f * acc0[r], 0.0f);
            float d1 = fmaxf(sn2[16 + M] + n2j - 2.0f * acc1[r], 0.0f);
            Dt[M][wave * 16 + l]      = d0;
            Dt[16 + M][wave * 16 + l] = d1;
        }
        __syncthreads();

        // Top-8 maintenance: thread r (< 32) owns row r; bestV sorted ascending.
        if (tid < ROWS_PER_BLK) {
            const int gi = r0 + tid;
#pragma unroll 1
            for (int c = 0; c < COLS_PER_TILE; ++c) {
                float v = Dt[tid][c];
                int   j = cT + c;
                if (v < bestV[KNN - 1] && j != gi) {
                    int pos = KNN - 1;
#pragma unroll
                    for (int p = KNN - 2; p >= 0; --p) {
                        if (v < bestV[p]) {
                            bestV[p + 1] = bestV[p];
                            bestI[p + 1] = bestI[p];
                            pos = p;
                        }
                    }
                    bestV[pos] = v;
                    bestI[pos] = j;
                }
            }
        }
        __syncthreads();
    }

    if (tid < ROWS_PER_BLK) {
        const int gi = r0 + tid;
#pragma unroll
        for (int k = 0; k < KNN; ++k) {
            nbr[gi * KNN + k]  = bestI[k];
            dval[gi * KNN + k] = bestV[k];
        }
    }
}

// One thread per directed edge.  Full-matrix sum counts each union pair twice:
// mutual edges appear twice in the directed list (weight 1), non-mutual once (weight 2).
__global__ __launch_bounds__(256) void gl_edge_loss(const float* __restrict__ Z,
                                                    const int* __restrict__ nbr,
                                                    const float* __restrict__ dval,
                                                    float* __restrict__ out) {
    __shared__ float red[256];
    const int e = blockIdx.x * 256 + threadIdx.x;
    const int i = e >> 3;
    const int j = nbr[e];
    const float w = expf(-dval[e]);   // SIGMA = 1

    const float4* Zi = reinterpret_cast<const float4*>(Z + (size_t)i * DZ);
    const float4* Zj = reinterpret_cast<const float4*>(Z + (size_t)j * DZ);
    float s = 0.0f;
#pragma unroll
    for (int d = 0; d < DZ / 4; ++d) {
        float4 a = Zi[d], b = Zj[d];
        float t0 = a.x - b.x, t1 = a.y - b.y, t2 = a.z - b.z, t3 = a.w - b.w;
        s += t0 * t0 + t1 * t1 + t2 * t2 + t3 * t3;
    }
    bool mutual = false;
#pragma unroll
    for (int m = 0; m < KNN; ++m) mutual = mutual || (nbr[j * KNN + m] == i);

    red[threadIdx.x] = w * s * (mutual ? 1.0f : 2.0f);
    __syncthreads();
    for (int st = 128; st >= 1; st >>= 1) {
        if (threadIdx.x < st) red[threadIdx.x] += red[threadIdx.x + st];
        __syncthreads();
    }
    if (threadIdx.x == 0)
        atomicAdd(out, red[0] * (1.0f / (float)(NPTS * KNN)));
}

extern "C" void kernel_launch(void* const* d_in, const int* in_sizes, int n_in,
                              void* d_out, int out_size, void* d_ws, size_t ws_size,
                              hipStream_t stream) {
    (void)in_sizes; (void)n_in; (void)out_size; (void)ws_size;
    const float* X = (const float*)d_in[0];   // 8192 x 512
    const float* Z = (const float*)d_in[1];   // 8192 x 64
    float* out = (float*)d_out;

    float* n2  = (float*)d_ws;                                        // 8192 floats
    int*   nbr = (int*)((char*)d_ws + (size_t)NPTS * 4);              // 8192*8 ints
    float* dvl = (float*)((char*)d_ws + (size_t)NPTS * 4
                                       + (size_t)NPTS * KNN * 4);     // 8192*8 floats

    hipLaunchKernelGGL(gl_zero_out,  dim3(1), dim3(1), 0, stream, out);
    hipLaunchKernelGGL(gl_row_norms, dim3(NPTS / 8), dim3(256), 0, stream, X, n2);
    hipLaunchKernelGGL(gl_knn_kernel, dim3(NPTS / ROWS_PER_BLK), dim3(128), 0, stream,
                       X, n2, nbr, dvl);
    hipLaunchKernelGGL(gl_edge_loss, dim3(NPTS * KNN / 256), dim3(256), 0, stream,
                       Z, nbr, dvl, out);
}